// MyDMPNN_54030688584200
// MI455X (gfx1250) — compile-verified
//
#include <hip/hip_runtime.h>

typedef float v2f __attribute__((ext_vector_type(2)));
typedef float v8f __attribute__((ext_vector_type(8)));
typedef int   v4i __attribute__((vector_size(4 * sizeof(int))));  // matches builtin proto

#define N_ATOMS   200000
#define N_BONDSP1 400001
#define MAX_NB    6
#define AF_DIM    133
#define FIN_DIM   147
#define DH        300
#define N_MOLS    1000
#define APM       200
#define GF_DIM    2048
#define OUT_COLS  (DH + GF_DIM)    // 2348
#define KMAX_PAD  448              // covers K=433 in 64-chunks
#define KCHUNK    64
#define BM        128              // M rows per block (8 wave32 x 16)
#define NTILES    ((DH + 15) / 16) // 19

#if __has_builtin(__builtin_amdgcn_global_load_async_to_lds_b32) && \
    __has_builtin(__builtin_amdgcn_global_load_async_to_lds_b128) && \
    __has_builtin(__builtin_amdgcn_s_wait_asynccnt)
#define HAVE_ASYNC_LDS 1
#else
#define HAVE_ASYNC_LDS 0
#endif

// Exact-match pointer casts for the async-to-LDS builtins (AS1 global src,
// AS3 LDS dst; C-style cast performs const_cast + addrspacecast).
#define GP128(p) ((__attribute__((address_space(1))) v4i*)(p))
#define GP32(p)  ((__attribute__((address_space(1))) int*)(p))
#define LP128(p) ((__attribute__((address_space(3))) v4i*)(p))
#define LP32(p)  ((__attribute__((address_space(3))) int*)(p))

// MODE 0: outPre = A*W^T ; outAct = relu(outPre)                      (W_i pass, K=147)
// MODE 1: outAct = relu(addsrc + A*W^T)                               (W_h pass, K=300)
// MODE 2: A = concat(A0[133], A1[300]); outAct = relu(A*W^T + bias)   (W_o pass, K=433)
template <int MODE>
__global__ __launch_bounds__(256) void gemm_relu_wmma(
    const float* __restrict__ A0, const float* __restrict__ A1,
    const float* __restrict__ Bw,        // [DH][K] row-major
    const float* __restrict__ addsrc,    // MODE 1
    const float* __restrict__ bias,      // MODE 2
    float* __restrict__ outAct, float* __restrict__ outPre,
    int M, int K, int splitK) {
  __shared__ float Bsh[KMAX_PAD][16];         // 28.7 KB weight column panel
  __shared__ float Ash[2][BM][KCHUNK + 4];    // 2 x 34.8 KB double-buffered A

  const int tid  = threadIdx.x;
  const int lane = tid & 31;
  const int wave = tid >> 5;
  const int m0   = blockIdx.x * BM;
  const int n0   = blockIdx.y * 16;
  const int nChunks = (K + KCHUNK - 1) / KCHUNK;
  const int Kpad    = nChunks * KCHUNK;

  // Stage B column tile [Kpad x 16]; ZERO-pad K tail and N>=DH columns.
  // These zeros make clamped (garbage) A pad values harmless: a * 0 == 0.
  for (int i = tid; i < Kpad * 16; i += 256) {
    int k = i >> 4, j = i & 15;
    float v = 0.f;
    if (k < K && (n0 + j) < DH) v = Bw[(long)(n0 + j) * K + k];
    Bsh[k][j] = v;
  }

  // Per-wave async-issue count per stage (compile-time constant: all issues
  // are unconditional with clamped addresses, so EXEC is always all-ones).
  constexpr int ISSUES = (MODE == 1) ? 8 : 32;

  auto stageA = [&](int kb, float(*dst)[KCHUNK + 4]) {
    if (MODE == 1) {  // 16B-aligned rows (K=300 -> 1200B stride), b128 copies
#pragma unroll
      for (int it = 0; it < 8; ++it) {
        int i  = tid + it * 256;          // [0, 2048)
        int r  = i >> 4;
        int kc = (i & 15) << 2;
        int gr = m0 + r;  gr = (gr < M) ? gr : (M - 1);
        int gk = kb + kc; gk = (gk <= K - 4) ? gk : (K - 4);
        const float* src = A0 + (long)gr * K + gk;
#if HAVE_ASYNC_LDS
        __builtin_amdgcn_global_load_async_to_lds_b128(GP128(src),
                                                       LP128(&dst[r][kc]), 0, 0);
#else
        *(float4*)&dst[r][kc] = *(const float4*)src;
#endif
      }
    } else {  // 4B-aligned rows / concat source, b32 copies
#pragma unroll
      for (int it = 0; it < 32; ++it) {
        int i  = tid + it * 256;          // [0, 8192)
        int r  = i >> 6;
        int kk = i & 63;
        int gr = m0 + r;  gr = (gr < M) ? gr : (M - 1);
        int gk = kb + kk; gk = (gk < K) ? gk : (K - 1);
        const float* src;
        if (MODE == 2) {
          src = (gk < splitK) ? A0 + (long)gr * splitK + gk
                              : A1 + (long)gr * DH + (gk - splitK);
        } else {
          src = A0 + (long)gr * K + gk;
        }
#if HAVE_ASYNC_LDS
        __builtin_amdgcn_global_load_async_to_lds_b32(GP32(src),
                                                      LP32(&dst[r][kk]), 0, 0);
#else
        dst[r][kk] = *src;
#endif
      }
    }
  };

  v8f acc = {0.f, 0.f, 0.f, 0.f, 0.f, 0.f, 0.f, 0.f};
  const int khalf = (lane < 16) ? 0 : 2;      // A/B K-pair per half-wave
  const int mloc  = wave * 16 + (lane & 15);  // A fragment row in block tile

  stageA(0, Ash[0]);  // prologue: first A chunk in flight

  for (int ch = 0; ch < nChunks; ++ch) {
    const int cur = ch & 1;
    const int kb  = ch * KCHUNK;
    if (ch + 1 < nChunks) {
      stageA(kb + KCHUNK, Ash[cur ^ 1]);  // issue next chunk into other buffer
#if HAVE_ASYNC_LDS
      // in-order completion: <=ISSUES outstanding ==> chunk `ch` has landed
      __builtin_amdgcn_s_wait_asynccnt(ISSUES);
#endif
    } else {
#if HAVE_ASYNC_LDS
      __builtin_amdgcn_s_wait_asynccnt(0);
#endif
    }
    __syncthreads();

#pragma unroll
    for (int kk = 0; kk < KCHUNK; kk += 4) {
      v2f a, b;
      a.x = Ash[cur][mloc][kk + khalf];
      a.y = Ash[cur][mloc][kk + khalf + 1];
      b.x = Bsh[kb + kk + khalf][lane & 15];
      b.y = Bsh[kb + kk + khalf + 1][lane & 15];
      acc = __builtin_amdgcn_wmma_f32_16x16x4_f32(false, a, false, b,
                                                  (short)0, acc, false, false);
    }
    __syncthreads();  // protect buffer `cur` before chunk ch+2 overwrites it
  }

  // C/D layout: VGPR v -> M = v (lanes 0-15) / v+8 (lanes 16-31); N = lane%16.
  const int col = n0 + (lane & 15);
#pragma unroll
  for (int v = 0; v < 8; ++v) {
    int row = m0 + wave * 16 + v + ((lane < 16) ? 0 : 8);
    if (row < M && col < DH) {
      float d = acc[v];
      long  o = (long)row * DH + col;
      if (MODE == 0) {
        outPre[o] = d;
        outAct[o] = d > 0.f ? d : 0.f;
      } else if (MODE == 1) {
        float s = addsrc[o] + d;
        outAct[o] = s > 0.f ? s : 0.f;
      } else {
        float s = d + bias[col];
        outAct[o] = s > 0.f ? s : 0.f;
      }
    }
  }
}

// dst[row] = sum_{j<6} src[idx[row][j]]  (rowlen DH, float4 per thread per neighbor)
__global__ void gather_sum6(const float* __restrict__ src,
                            const int* __restrict__ idx,
                            float* __restrict__ dst, int nrows) {
  const int chunks = DH / 4;  // 75
  long t = (long)blockIdx.x * blockDim.x + threadIdx.x;
  if (t >= (long)nrows * chunks) return;
  int row = (int)(t / chunks);
  int c   = (int)(t - (long)row * chunks) * 4;
  const int* ir = idx + (long)row * MAX_NB;
  float4 s = make_float4(0.f, 0.f, 0.f, 0.f);
#pragma unroll
  for (int j = 0; j < MAX_NB; ++j) {
    const float4 v = *(const float4*)(src + (long)ir[j] * DH + c);
    s.x += v.x; s.y += v.y; s.z += v.z; s.w += v.w;
  }
  *(float4*)(dst + (long)row * DH + c) = s;
}

// out[mol][0:300]   = mean over 200 atoms of atoms_h
// out[mol][300:2348]= global_features[mol]
__global__ void readout_kernel(const float* __restrict__ atoms_h,
                               const float* __restrict__ gfeat,
                               float* __restrict__ out) {
  long t = (long)blockIdx.x * blockDim.x + threadIdx.x;
  if (t >= (long)N_MOLS * OUT_COLS) return;
  int mol = (int)(t / OUT_COLS);
  int col = (int)(t - (long)mol * OUT_COLS);
  if (col < DH) {
    float s = 0.f;
    const float* p = atoms_h + (long)mol * APM * DH + col;
#pragma unroll 4
    for (int i = 0; i < APM; ++i) s += p[(long)i * DH];
    out[t] = s * (1.f / APM);
  } else {
    out[t] = gfeat[(long)mol * GF_DIM + (col - DH)];
  }
}

extern "C" void kernel_launch(void* const* d_in, const int* in_sizes, int n_in,
                              void* d_out, int out_size, void* d_ws, size_t ws_size,
                              hipStream_t stream) {
  const float* atom_features = (const float*)d_in[0];
  const float* f_bonds       = (const float*)d_in[1];
  const int*   a2b           = (const int*)d_in[2];
  const int*   mapping       = (const int*)d_in[3];
  const float* gfeat         = (const float*)d_in[4];
  const float* W_i           = (const float*)d_in[5];
  const float* W_h           = (const float*)d_in[6];
  const float* W_o_w         = (const float*)d_in[7];
  const float* W_o_b         = (const float*)d_in[8];

  float* ws = (float*)d_ws;
  const long SZB = (long)N_BONDSP1 * DH;
  const long SZA = (long)N_ATOMS * DH;
  float* inp     = ws;                  // pre-activation bond features
  float* message = ws + SZB;            // current bond messages
  float* mbuf    = ws + 2 * SZB;        // gathered neighbor sums
  float* mta     = ws + 3 * SZB;        // messages aggregated to atoms
  float* atoms_h = ws + 3 * SZB + SZA;  // atom hidden states

  dim3 blk(256);
  dim3 gB((N_BONDSP1 + BM - 1) / BM, NTILES);
  dim3 gA((N_ATOMS + BM - 1) / BM, NTILES);

  // inp = f_bonds @ W_i^T ; message = relu(inp)
  gemm_relu_wmma<0><<<gB, blk, 0, stream>>>(f_bonds, nullptr, W_i, nullptr, nullptr,
                                            message, inp, N_BONDSP1, FIN_DIM, 0);

  long gth = (long)N_BONDSP1 * (DH / 4);
  dim3 gGb((unsigned)((gth + 255) / 256));
  for (int d = 1; d < 3; ++d) {  // DEPTH = 3
    gather_sum6<<<gGb, blk, 0, stream>>>(message, mapping, mbuf, N_BONDSP1);
    gemm_relu_wmma<1><<<gB, blk, 0, stream>>>(mbuf, nullptr, W_h, inp, nullptr,
                                              message, nullptr, N_BONDSP1, DH, 0);
  }

  long ath = (long)N_ATOMS * (DH / 4);
  dim3 gGa((unsigned)((ath + 255) / 256));
  gather_sum6<<<gGa, blk, 0, stream>>>(message, a2b, mta, N_ATOMS);

  // atoms_h = relu([atom_features | mta] @ W_o^T + b)
  gemm_relu_wmma<2><<<gA, blk, 0, stream>>>(atom_features, mta, W_o_w, nullptr, W_o_b,
                                            atoms_h, nullptr, N_ATOMS, AF_DIM + DH, AF_DIM);

  long oth = (long)N_MOLS * OUT_COLS;
  readout_kernel<<<(unsigned)((oth + 255) / 256), blk, 0, stream>>>(atoms_h, gfeat,
                                                                    (float*)d_out);
}